// De_conv_batched_multimasks_7627861917769
// MI455X (gfx1250) — compile-verified
//
#include <hip/hip_runtime.h>
#include <stdint.h>

// ---------------------------------------------------------------------------
// De_conv_batched_multimasks — MI455X (gfx1250) implementation.
//
// Per area (16x16 tile): diff_round chain -> mc, soft erosion (neighbors via
// LDS), masked edges out, 4 tile reductions -> scalar out.
//
// CDNA5 specifics used:
//   * v_sin_f32 via __builtin_amdgcn_sinf: hardware computes sin(2*pi*x),
//     exactly the diff_round term -> 1 trans op per diff_round, no scaling.
//   * global_load_async_to_lds_b128 + s_wait_asynccnt: async staging of the
//     6KB area tile into LDS (ASYNCcnt path, CDNA5 ISA).
//   * wave32 shfl_xor reductions (warpSize==32).
//   * hdr(mid) computed once per area by 4 lanes (saves ~27% of all
//     v_sin_f32 work vs per-thread recompute), published via LDS using the
//     barrier already required for the async staging.
//   * non-temporal store of the streamed masked_edges output.
// ---------------------------------------------------------------------------

#define INV_TWO_PI 0.15915494309189535f
#define EPSV 1e-8f

// diff_round(x) = x - sin(2*pi*x)/(2*pi).  V_SIN_F32 input is in rotations:
// __builtin_amdgcn_sinf(x) == sin(2*pi*x)  (valid for |x| <= 256; our values
// stay in ~[-0.05, 1.05]).
__device__ __forceinline__ float dr(float x) {
    return x - INV_TWO_PI * __builtin_amdgcn_sinf(x);
}
__device__ __forceinline__ float hdr(float x) { return dr(dr(x)); }

// Generic pointers to __shared__ on gfx1250 are {shared_aperture_hi, lds_off};
// low 32 bits are the wave-relative LDS byte address (ISA 10.2 aperture map).
__device__ __forceinline__ uint32_t lds_off(const void* p) {
    return (uint32_t)(uintptr_t)p;
}

// Async DMA of 16 bytes global -> LDS. Tracked by ASYNCcnt.
__device__ __forceinline__ void async_copy_b128(uint32_t lds_addr, const void* gaddr) {
    asm volatile("global_load_async_to_lds_b128 %0, %1, off"
                 :: "v"(lds_addr), "v"(gaddr)
                 : "memory");
}
__device__ __forceinline__ void wait_asynccnt0() {
    asm volatile("s_wait_asynccnt 0x0" ::: "memory");
}

__global__ __launch_bounds__(256) void de_conv_areas_kernel(
    const float* __restrict__ img,    // [N,16,16,1]
    const float* __restrict__ mask,   // [N,16,16,4]
    const float* __restrict__ mid,    // [N,4]
    const float* __restrict__ edge,   // [N,16,16]
    float* __restrict__ out_me,       // [N,16,16,1]
    float* __restrict__ out_s,        // [N]
    int n_areas)
{
    __shared__ __align__(16) float4 s_mask[256];   // 4 KB
    __shared__ __align__(16) float  s_img[256];    // 1 KB
    __shared__ __align__(16) float  s_edge[256];   // 1 KB
    __shared__ float s_mc[16][16];                 // 1 KB
    __shared__ float s_part[8][4];                 // wave partials
    __shared__ float s_midb[4];                    // hdr(mid[c]) per channel

    const int t    = threadIdx.x;
    const int area = blockIdx.x;
    if (area >= n_areas) return;   // uniform per block

    const float* gmask = mask + (size_t)area * 1024;
    const float* gimg  = img  + (size_t)area * 256;
    const float* gedge = edge + (size_t)area * 256;

    // ---- async stage: 6 KB tile -> LDS --------------------------------
    // mask: 256 threads x 16B; img: threads 0..63 x 16B; edge: 64..127 x 16B
    async_copy_b128(lds_off(&s_mask[t]), gmask + (size_t)t * 4);
    if (t < 64) {
        async_copy_b128(lds_off(&s_img[t * 4]), gimg + (size_t)t * 4);
    } else if (t < 128) {
        async_copy_b128(lds_off(&s_edge[(t - 64) * 4]), gedge + (size_t)(t - 64) * 4);
    }

    // Overlap with the DMA: harder_diff_round(mid) once per area (4 lanes),
    // published via LDS through the barrier below.
    if (t < 4) {
        s_midb[t] = hdr(mid[(size_t)area * 4 + t]);
    }

    wait_asynccnt0();       // this wave's async copies landed in LDS
    __syncthreads();        // all waves' copies + s_midb visible block-wide

    const float4 m4 = s_mask[t];
    const float  ip = s_img[t];
    const float  ep = s_edge[t];
    const float  b0 = s_midb[0];
    const float  b1 = s_midb[1];
    const float  b2 = s_midb[2];
    const float  b3 = s_midb[3];

    // differentiable_eq(mask, mid) with hdr(mid) hoisted
    auto eq = [](float mv, float bb) {
        const float a = hdr(mv);
        return hdr(a * bb + (1.f - a) * (1.f - bb));
    };
    const float e0 = eq(m4.x, b0);
    const float e1 = eq(m4.y, b1);
    const float e2 = eq(m4.z, b2);
    const float e3 = eq(m4.w, b3);
    const float pa = dr(e0) * dr(e1);          // differentiable_and
    const float pb = dr(e2) * dr(e3);
    const float mc = dr(pa) * dr(pb);          // [16,16] soft mask value
    const float mi = mc * ip;                  // masked_image

    const int r = t >> 4, c = t & 15;
    s_mc[r][c] = mc;

    // ---- reduction pass A: sum(mc), sum(mc*img) -----------------------
    float smc = mc, smi = mi;
    #pragma unroll
    for (int off = 16; off; off >>= 1) {
        smc += __shfl_xor(smc, off, 32);
        smi += __shfl_xor(smi, off, 32);
    }
    const int wave = t >> 5, lane = t & 31;
    if (lane == 0) { s_part[wave][0] = smc; s_part[wave][1] = smi; }
    __syncthreads();   // also publishes s_mc for the erosion neighbors

    float sum_mc = 0.f, sum_mi = 0.f;
    #pragma unroll
    for (int w = 0; w < 8; ++w) { sum_mc += s_part[w][0]; sum_mi += s_part[w][1]; }
    const float denom = sum_mc + EPSV;
    const float meann = sum_mi / denom;

    // ---- soft erosion (zero-padded shifts of mc) ----------------------
    const float prev = (r < 15) ? s_mc[r + 1][c] : 0.f;  // translate(axis0, bwd)
    const float nxt  = (r > 0)  ? s_mc[r - 1][c] : 0.f;  // translate(axis0, fwd)
    const float up   = (c < 15) ? s_mc[r][c + 1] : 0.f;  // translate(axis1, bwd)
    const float down = (c > 0)  ? s_mc[r][c - 1] : 0.f;  // translate(axis1, fwd)

    const float ed   = 1.f - (prev * nxt + (1.f - prev) * (1.f - nxt));
    const float edud = 1.f - (up * down + (1.f - up) * (1.f - down));
    const float s0a  = ed * mc;
    const float s0b  = edud * mc;
    const float sb0  = s0a * s0b + (1.f - s0a) * s0a + (1.f - s0b) * s0a; // or_simple
    const float eroded = (1.f - sb0) * mc;
    const float me = eroded * ep;              // masked_edges

    // Streamed output, never re-read: keep it out of the cache hierarchy.
    __builtin_nontemporal_store(me, out_me + (size_t)area * 256 + t);

    // ---- reduction pass B: sum(me), sum(((mi-mean)*mc)^2) -------------
    float vt = (mi - meann) * mc;
    vt *= vt;
    float sme = me, sv = vt;
    #pragma unroll
    for (int off = 16; off; off >>= 1) {
        sme += __shfl_xor(sme, off, 32);
        sv  += __shfl_xor(sv,  off, 32);
    }
    if (lane == 0) { s_part[wave][2] = sme; s_part[wave][3] = sv; }
    __syncthreads();

    if (t == 0) {
        float sum_me = 0.f, sum_v = 0.f;
        #pragma unroll
        for (int w = 0; w < 8; ++w) { sum_me += s_part[w][2]; sum_v += s_part[w][3]; }
        const float loss = sum_me * (1.f / 256.f);   // mean(masked_edges)
        const float varr = sum_v / denom;
        out_s[area] = varr * loss * 1000.f;
    }
}

extern "C" void kernel_launch(void* const* d_in, const int* in_sizes, int n_in,
                              void* d_out, int out_size, void* d_ws, size_t ws_size,
                              hipStream_t stream) {
    (void)n_in; (void)d_ws; (void)ws_size; (void)out_size;
    const float* img  = (const float*)d_in[0];   // resized_image  [B,A,16,16,1]
    const float* mask = (const float*)d_in[1];   // mask_combined  [B,A,16,16,4]
    const float* mid  = (const float*)d_in[2];   // initial_mask_id [B,A,4]
    const float* edge = (const float*)d_in[3];   // edge_map       [B,A,16,16]

    const int n_areas = in_sizes[2] / 4;         // B*A = 16384

    float* out_me = (float*)d_out;                          // [B,A,16,16,1]
    float* out_s  = (float*)d_out + (size_t)n_areas * 256;  // [B,A]

    de_conv_areas_kernel<<<n_areas, 256, 0, stream>>>(
        img, mask, mid, edge, out_me, out_s, n_areas);
}